// NoisyLabelEmbedding_56710748176534
// MI455X (gfx1250) — compile-verified
//
#include <hip/hip_runtime.h>

// ---------------------------------------------------------------------------
// NoisyLabelEmbedding for MI455X (gfx1250, wave32).
//
// Key algebraic facts exploited (exact, not approximations):
//  * semantic_diff = broadcast(sem) - sem == 0, so Wf1 only sees the 3
//    position channels (rows 64..66 of Wf1) -> semantic_condition unused.
//  * sum_k (w_k @ Wf2 + bf2) * x = (sum_k w_k) @ Wf2 * x + 16*bf2*x,
//    so the big 1M x 64 x 64 GEMM collapses to 65536 x 64 x 64.
//  * BN of an affine map: mean/var per column derived from first/second
//    moments of the raw inputs (104 scalars for NL, 9 for POS).
//
// Resulting traffic ~80 MB -> memory bound at 23.3 TB/s; f32 WMMA
// (v_wmma_f32_16x16x4_f32) keeps reference fp32 numerics.
// ---------------------------------------------------------------------------

typedef __attribute__((ext_vector_type(2))) float v2f;
typedef __attribute__((ext_vector_type(8))) float v8f;

#define ROWS_H    65536      // B*N
#define ROWS_W    1048576    // B*N*K
#define KNBR      16
#define CIN       13
#define COUT      64
#define NL_STATS  104        // 13 sums + 91 pair products
#define POS_STATS 9          // 3 sums + 6 pair products
#define RED_BLOCKS 64
#define BN_EPS 1e-5f

__device__ __forceinline__ v8f wmma4(v2f a, v2f b, v8f c) {
  // D = A(16x4,f32) x B(4x16,f32) + C(16x16,f32)
  return __builtin_amdgcn_wmma_f32_16x16x4_f32(
      /*neg_a=*/false, a, /*neg_b=*/false, b,
      /*c_mod=*/(short)0, c, /*reuse_a=*/false, /*reuse_b=*/false);
}

// ---------------------------------------------------------------------------
// Stage 1a: moments of noisy_labels (13 sums + 91 upper-tri products).
// ---------------------------------------------------------------------------
__global__ __launch_bounds__(256) void stats_nl_kernel(
    const float* __restrict__ nl, float* __restrict__ part) {
  float acc[NL_STATS];
#pragma unroll
  for (int i = 0; i < NL_STATS; ++i) acc[i] = 0.f;
  for (int r = blockIdx.x * blockDim.x + threadIdx.x; r < ROWS_H;
       r += gridDim.x * blockDim.x) {
    float row[CIN];
#pragma unroll
    for (int p = 0; p < CIN; ++p) row[p] = nl[r * CIN + p];
#pragma unroll
    for (int p = 0; p < CIN; ++p) acc[p] += row[p];
    int idx = CIN;
#pragma unroll
    for (int p = 0; p < CIN; ++p)
#pragma unroll
      for (int q = p; q < CIN; ++q) acc[idx++] += row[p] * row[q];
  }
  __shared__ float sPart[8 * NL_STATS];
  const int lane = threadIdx.x & 31, wave = threadIdx.x >> 5;
#pragma unroll
  for (int i = 0; i < NL_STATS; ++i) {
    float v = acc[i];
    v += __shfl_xor(v, 16, 32);
    v += __shfl_xor(v, 8, 32);
    v += __shfl_xor(v, 4, 32);
    v += __shfl_xor(v, 2, 32);
    v += __shfl_xor(v, 1, 32);
    if (lane == 0) sPart[wave * NL_STATS + i] = v;
  }
  __syncthreads();
  for (int t = threadIdx.x; t < NL_STATS; t += blockDim.x) {
    float s = 0.f;
    for (int w = 0; w < 8; ++w) s += sPart[w * NL_STATS + t];
    part[blockIdx.x * NL_STATS + t] = s;
  }
}

// ---------------------------------------------------------------------------
// Stage 1b: moments of position_condition (3 sums + 6 products).
// ---------------------------------------------------------------------------
__global__ __launch_bounds__(256) void stats_pos_kernel(
    const float* __restrict__ pos, float* __restrict__ part) {
  float acc[POS_STATS];
#pragma unroll
  for (int i = 0; i < POS_STATS; ++i) acc[i] = 0.f;
  for (int r = blockIdx.x * blockDim.x + threadIdx.x; r < ROWS_W;
       r += gridDim.x * blockDim.x) {
    const float x = pos[r * 3 + 0];
    const float y = pos[r * 3 + 1];
    const float z = pos[r * 3 + 2];
    acc[0] += x; acc[1] += y; acc[2] += z;
    acc[3] += x * x; acc[4] += x * y; acc[5] += x * z;
    acc[6] += y * y; acc[7] += y * z; acc[8] += z * z;
  }
  __shared__ float sPart[8 * POS_STATS];
  const int lane = threadIdx.x & 31, wave = threadIdx.x >> 5;
#pragma unroll
  for (int i = 0; i < POS_STATS; ++i) {
    float v = acc[i];
    v += __shfl_xor(v, 16, 32);
    v += __shfl_xor(v, 8, 32);
    v += __shfl_xor(v, 4, 32);
    v += __shfl_xor(v, 2, 32);
    v += __shfl_xor(v, 1, 32);
    if (lane == 0) sPart[wave * POS_STATS + i] = v;
  }
  __syncthreads();
  for (int t = threadIdx.x; t < POS_STATS; t += blockDim.x) {
    float s = 0.f;
    for (int w = 0; w < 8; ++w) s += sPart[w * POS_STATS + t];
    part[blockIdx.x * POS_STATS + t] = s;
  }
}

// ---------------------------------------------------------------------------
// Stage 2: fold moments into BN affine coefficients.
//   coef[0..63]    a1   (h path scale),   coef[64..127]  c1 (h path shift,
//                                          b1 folded in)
//   coef[128..191] af   (w path scale),   coef[192..255] cf (w path shift)
// ---------------------------------------------------------------------------
__global__ void finalize_kernel(
    const float* __restrict__ partNL, const float* __restrict__ partPOS,
    const float* __restrict__ W1, const float* __restrict__ b1,
    const float* __restrict__ g1, const float* __restrict__ be1,
    const float* __restrict__ Wf1, const float* __restrict__ bf1,
    const float* __restrict__ gf, const float* __restrict__ bef,
    float* __restrict__ coef) {
  __shared__ float S[NL_STATS];
  __shared__ float P[POS_STATS];
  const int t = threadIdx.x;
  if (t < NL_STATS) {
    float s = 0.f;
    for (int bkt = 0; bkt < RED_BLOCKS; ++bkt) s += partNL[bkt * NL_STATS + t];
    S[t] = s;
  } else if (t < NL_STATS + POS_STATS) {
    const int i = t - NL_STATS;
    float s = 0.f;
    for (int bkt = 0; bkt < RED_BLOCKS; ++bkt) s += partPOS[bkt * POS_STATS + i];
    P[i] = s;
  }
  __syncthreads();
  const int j = t;
  if (j < COUT) {
    // ---- h = NL@W1 + b1 statistics (analytic) ----
    const float invR = 1.f / (float)ROWS_H;
    float w[CIN], m[CIN];
#pragma unroll
    for (int p = 0; p < CIN; ++p) {
      w[p] = W1[p * COUT + j];
      m[p] = S[p] * invR;
    }
    float dotm = 0.f;
#pragma unroll
    for (int p = 0; p < CIN; ++p) dotm = fmaf(m[p], w[p], dotm);
    const float b1j = b1[j];
    const float mu = b1j + dotm;
    float e2 = fmaf(b1j, b1j, 2.f * b1j * dotm);
    int idx = CIN;
#pragma unroll
    for (int p = 0; p < CIN; ++p)
#pragma unroll
      for (int q = p; q < CIN; ++q) {
        const float Mpq = S[idx++] * invR;
        const float f = (p == q) ? 1.f : 2.f;
        e2 = fmaf(f * Mpq, w[p] * w[q], e2);
      }
    const float var = e2 - mu * mu;
    const float a1 = g1[j] * rsqrtf(var + BN_EPS);
    coef[j] = a1;
    coef[64 + j] = be1[j] - a1 * dotm;  // b1 folded in (GEMM omits bias)

    // ---- w = pos@Wf1[64:67] + bf1 statistics (analytic) ----
    const float invRp = 1.f / (float)ROWS_W;
    float wf[3], mp[3];
#pragma unroll
    for (int i = 0; i < 3; ++i) {
      wf[i] = Wf1[(COUT + i) * COUT + j];
      mp[i] = P[i] * invRp;
    }
    float dotf = 0.f;
#pragma unroll
    for (int i = 0; i < 3; ++i) dotf = fmaf(mp[i], wf[i], dotf);
    const float bfj = bf1[j];
    const float muf = bfj + dotf;
    float e2f = fmaf(bfj, bfj, 2.f * bfj * dotf);
    idx = 3;
#pragma unroll
    for (int i = 0; i < 3; ++i)
#pragma unroll
      for (int q = i; q < 3; ++q) {
        const float Mq = P[idx++] * invRp;
        const float f = (i == q) ? 1.f : 2.f;
        e2f = fmaf(f * Mq, wf[i] * wf[q], e2f);
      }
    const float varf = e2f - muf * muf;
    const float af = gf[j] * rsqrtf(varf + BN_EPS);
    coef[128 + j] = af;
    coef[192 + j] = bef[j] - af * muf;  // bias applied in kernel -> full shift
  }
}

// ---------------------------------------------------------------------------
// Stage 3: x = relu(a1*(NL@W1) + c1) @ W2 + b2.  One 16-row tile per wave.
// GEMM1: K padded 13->16 (4 WMMA K-steps), GEMM2: K=64 (16 K-steps), all
// v_wmma_f32_16x16x4_f32.  LDS tiles padded to stride 65 (bank-conflict free
// column reads on 64-bank LDS).
// ---------------------------------------------------------------------------
__global__ __launch_bounds__(256) void gemm_x_kernel(
    const float* __restrict__ nl, const float* __restrict__ W1,
    const float* __restrict__ W2, const float* __restrict__ b2,
    const float* __restrict__ coef, float* __restrict__ xbuf) {
  __shared__ float sW1[16 * 65];
  __shared__ float sW2[64 * 65];
  __shared__ float sA1[64], sC1[64], sB2[64];
  __shared__ float sH[8][16 * 65];
  const int tid = threadIdx.x;
  for (int i = tid; i < 16 * 64; i += 256) {
    const int p = i >> 6, j = i & 63;
    sW1[p * 65 + j] = (p < CIN) ? W1[p * 64 + j] : 0.f;  // zero-pad rows 13..15
  }
  for (int i = tid; i < 64 * 64; i += 256) {
    const int p = i >> 6, j = i & 63;
    sW2[p * 65 + j] = W2[i];
  }
  if (tid < 64) { sA1[tid] = coef[tid]; sC1[tid] = coef[64 + tid]; sB2[tid] = b2[tid]; }
  __syncthreads();

  const int wave = tid >> 5, lane = tid & 31;
  const int half = lane >> 4, mrow = lane & 15;
  const int r0 = (blockIdx.x * 8 + wave) * 16;
  float* hT = &sH[wave][0];

  const v8f z = {0.f, 0.f, 0.f, 0.f, 0.f, 0.f, 0.f, 0.f};
  v8f acc[4] = {z, z, z, z};
#pragma unroll
  for (int kk = 0; kk < 16; kk += 4) {           // GEMM1: h = NLpad @ W1pad
    const int c0 = kk + 2 * half;                // A layout: K = 2*(l/16)+v
    v2f a;
    a.x = (c0 < CIN)     ? nl[(r0 + mrow) * CIN + c0]     : 0.f;
    a.y = (c0 + 1 < CIN) ? nl[(r0 + mrow) * CIN + c0 + 1] : 0.f;
#pragma unroll
    for (int nt = 0; nt < 4; ++nt) {
      const int n = nt * 16 + mrow;
      v2f b;
      b.x = sW1[c0 * 65 + n];
      b.y = sW1[(c0 + 1) * 65 + n];
      acc[nt] = wmma4(a, b, acc[nt]);
    }
  }
  // BN affine + ReLU (C layout: row = v + 8*half, col = nt*16 + mrow) -> LDS
#pragma unroll
  for (int nt = 0; nt < 4; ++nt) {
    const int j = nt * 16 + mrow;
    const float a1j = sA1[j], c1j = sC1[j];
#pragma unroll
    for (int v = 0; v < 8; ++v) {
      const int M = v + 8 * half;
      hT[M * 65 + j] = fmaxf(fmaf(a1j, acc[nt][v], c1j), 0.f);
    }
  }
  v8f xacc[4] = {z, z, z, z};
#pragma unroll
  for (int kk = 0; kk < 64; kk += 4) {           // GEMM2: x = h @ W2
    const int k0 = kk + 2 * half;
    v2f a;
    a.x = hT[mrow * 65 + k0];
    a.y = hT[mrow * 65 + k0 + 1];
#pragma unroll
    for (int nt = 0; nt < 4; ++nt) {
      const int n = nt * 16 + mrow;
      v2f b;
      b.x = sW2[k0 * 65 + n];
      b.y = sW2[(k0 + 1) * 65 + n];
      xacc[nt] = wmma4(a, b, xacc[nt]);
    }
  }
#pragma unroll
  for (int nt = 0; nt < 4; ++nt) {
    const int j = nt * 16 + mrow;
    const float b2j = sB2[j];
#pragma unroll
    for (int v = 0; v < 8; ++v) {
      const int row = r0 + v + 8 * half;
      xbuf[row * 64 + j] = xacc[nt][v] + b2j;
    }
  }
}

// ---------------------------------------------------------------------------
// Stage 4 (fused): u_r = sum_k relu(af*(pos_rk@Wf1'+bf1)+cf)   (VALU, rank-3)
//                  s   = u @ Wf2                               (WMMA)
//                  y   = x * (s + 16*bf2 + 1)                  (fused epilogue)
// ---------------------------------------------------------------------------
__global__ __launch_bounds__(256) void fuse_w_kernel(
    const float* __restrict__ pos, const float* __restrict__ Wf1,
    const float* __restrict__ bf1, const float* __restrict__ Wf2,
    const float* __restrict__ bf2, const float* __restrict__ coef,
    const float* __restrict__ xbuf, float* __restrict__ out) {
  __shared__ float sWf2[64 * 65];
  __shared__ float sU[8][16 * 65];
  __shared__ float sWp[3][64], sBf1[64], sAf[64], sCf[64], sBf2[64];
  const int tid = threadIdx.x;
  for (int i = tid; i < 64 * 64; i += 256) {
    const int p = i >> 6, j = i & 63;
    sWf2[p * 65 + j] = Wf2[i];
  }
  if (tid < 64) {
    sWp[0][tid] = Wf1[64 * 64 + tid];   // position rows of Wf1 (67 x 64)
    sWp[1][tid] = Wf1[65 * 64 + tid];
    sWp[2][tid] = Wf1[66 * 64 + tid];
    sBf1[tid] = bf1[tid];
    sAf[tid] = coef[128 + tid];
    sCf[tid] = coef[192 + tid];
    sBf2[tid] = bf2[tid];
  }
  __syncthreads();

  const int wave = tid >> 5, lane = tid & 31;
  const int half = lane >> 4, mrow = lane & 15;
  const int r0 = (blockIdx.x * 8 + wave) * 16;
  float* uT = &sU[wave][0];

  const int j0 = lane, j1 = lane + 32;           // each lane owns 2 channels
  const float w00 = sWp[0][j0], w01 = sWp[0][j1];
  const float w10 = sWp[1][j0], w11 = sWp[1][j1];
  const float w20 = sWp[2][j0], w21 = sWp[2][j1];
  const float bb0 = sBf1[j0], bb1 = sBf1[j1];
  const float af0 = sAf[j0], af1 = sAf[j1];
  const float cf0 = sCf[j0], cf1 = sCf[j1];

  for (int rr = 0; rr < 16; ++rr) {
    const float* p = pos + (size_t)(r0 + rr) * (KNBR * 3);
    float a0 = 0.f, a1 = 0.f;
#pragma unroll
    for (int k = 0; k < KNBR; ++k) {
      const float px = p[3 * k], py = p[3 * k + 1], pz = p[3 * k + 2];
      const float t0 = fmaf(px, w00, fmaf(py, w10, fmaf(pz, w20, bb0)));
      const float t1 = fmaf(px, w01, fmaf(py, w11, fmaf(pz, w21, bb1)));
      a0 += fmaxf(fmaf(af0, t0, cf0), 0.f);
      a1 += fmaxf(fmaf(af1, t1, cf1), 0.f);
    }
    uT[rr * 65 + j0] = a0;
    uT[rr * 65 + j1] = a1;
  }

  const v8f z = {0.f, 0.f, 0.f, 0.f, 0.f, 0.f, 0.f, 0.f};
  v8f sacc[4] = {z, z, z, z};
#pragma unroll
  for (int kk = 0; kk < 64; kk += 4) {           // s = u @ Wf2
    const int k0 = kk + 2 * half;
    v2f a;
    a.x = uT[mrow * 65 + k0];
    a.y = uT[mrow * 65 + k0 + 1];
#pragma unroll
    for (int nt = 0; nt < 4; ++nt) {
      const int n = nt * 16 + mrow;
      v2f b;
      b.x = sWf2[k0 * 65 + n];
      b.y = sWf2[(k0 + 1) * 65 + n];
      sacc[nt] = wmma4(a, b, sacc[nt]);
    }
  }
#pragma unroll
  for (int nt = 0; nt < 4; ++nt) {
    const int j = nt * 16 + mrow;
    const float addc = 16.f * sBf2[j] + 1.f;     // y = x*(s + 16*bf2) + x
#pragma unroll
    for (int v = 0; v < 8; ++v) {
      const int row = r0 + v + 8 * half;
      const float xv = xbuf[row * 64 + j];
      out[row * 64 + j] = xv * (sacc[nt][v] + addc);
    }
  }
}

// ---------------------------------------------------------------------------
extern "C" void kernel_launch(void* const* d_in, const int* in_sizes, int n_in,
                              void* d_out, int out_size, void* d_ws,
                              size_t ws_size, hipStream_t stream) {
  (void)in_sizes; (void)n_in; (void)out_size; (void)ws_size;
  const float* nl  = (const float*)d_in[0];
  // d_in[1] = semantic_condition: unused (semantic_diff == 0 identically)
  const float* pos = (const float*)d_in[2];
  const float* W1  = (const float*)d_in[3];
  const float* b1  = (const float*)d_in[4];
  const float* g1  = (const float*)d_in[5];
  const float* be1 = (const float*)d_in[6];
  const float* W2  = (const float*)d_in[7];
  const float* b2  = (const float*)d_in[8];
  const float* Wf1 = (const float*)d_in[9];
  const float* bf1 = (const float*)d_in[10];
  const float* gf  = (const float*)d_in[11];
  const float* bef = (const float*)d_in[12];
  const float* Wf2 = (const float*)d_in[13];
  const float* bf2 = (const float*)d_in[14];
  float* out = (float*)d_out;

  float* ws      = (float*)d_ws;
  float* partNL  = ws;                                   // 64*104 floats
  float* partPOS = partNL + RED_BLOCKS * NL_STATS;       // 64*9 floats
  float* coef    = partPOS + RED_BLOCKS * POS_STATS;     // 256 floats
  float* xbuf    = coef + 256;                           // 65536*64 floats

  stats_nl_kernel<<<RED_BLOCKS, 256, 0, stream>>>(nl, partNL);
  stats_pos_kernel<<<RED_BLOCKS, 256, 0, stream>>>(pos, partPOS);
  finalize_kernel<<<1, 128, 0, stream>>>(partNL, partPOS, W1, b1, g1, be1,
                                         Wf1, bf1, gf, bef, coef);
  gemm_x_kernel<<<ROWS_H / 128, 256, 0, stream>>>(nl, W1, W2, b2, coef, xbuf);
  fuse_w_kernel<<<ROWS_H / 128, 256, 0, stream>>>(pos, Wf1, bf1, Wf2, bf2,
                                                  coef, xbuf, out);
}